// Memory_45569603010687
// MI455X (gfx1250) — compile-verified
//
#include <hip/hip_runtime.h>
#include <hip/hip_bf16.h>

// ---------------------------------------------------------------------------
// Problem constants (from reference): B=16, N=2000, D=C*H*W=512*7*7=25088
// ---------------------------------------------------------------------------
#define BB      16
#define NN      2000
#define DD      25088
#define PI_2F   (3.14159f * 0.5f)
#define EPS_SUM 1e-07f
#define EPS_COS 1e-08f

#define KB      128            // K-block staged in LDS per stage (kernel1)
#define LDA     (KB + 4)       // padded LDS row stride (132 % 64 == 4 -> conflict-free)
#define KSPLIT  7              // K split across blockIdx.y in kernel1
#define KCHUNK  (DD / KSPLIT)  // 3584 = 28 * KB
#define NTILEK  (KCHUNK / KB)  // 28 double-buffered stages

typedef float v2f __attribute__((ext_vector_type(2)));
typedef float v8f __attribute__((ext_vector_type(8)));

// D = A(16x4 f32) * B(4x16 f32) + C(16x16 f32), full-precision matrix core op.
__device__ __forceinline__ v8f wmma_f32(v2f a, v2f b, v8f c) {
    return __builtin_amdgcn_wmma_f32_16x16x4_f32(
        /*neg_a=*/false, a, /*neg_b=*/false, b,
        /*c_mod=*/(short)0, c, /*reuse_a=*/false, /*reuse_b=*/false);
}

// generic (__shared__) pointer -> raw 32-bit LDS byte offset for async-DMA ops
__device__ __forceinline__ uint32_t lds_offset(const void* p) {
    return (uint32_t)(__SIZE_TYPE__)(__attribute__((address_space(3))) const void*)p;
}

// CDNA5 async direct global->LDS copy (ASYNCcnt-tracked, bypasses VGPRs).
__device__ __forceinline__ void async_ld_b128(uint32_t lds_byte, const float* g) {
    asm volatile("global_load_async_to_lds_b128 %0, %1, off"
                 :: "v"(lds_byte), "v"(g) : "memory");
}
__device__ __forceinline__ void wait_async0() {
    asm volatile("s_wait_asynccnt 0x0" ::: "memory");
}

// ---------------------------------------------------------------------------
// Kernel 0: per-key sum & sumsq -> knorm[b][d] = key/(sum+eps), kn[b]=||knorm||
// ---------------------------------------------------------------------------
__global__ __launch_bounds__(256) void k0_normalize(
        const float* __restrict__ key, float* __restrict__ knorm,
        float* __restrict__ kn) {
    int b = blockIdx.x, tid = threadIdx.x;
    const float* kp = key + (size_t)b * DD;
    float s = 0.f, sq = 0.f;
    for (int i = tid; i < DD; i += 256) { float v = kp[i]; s += v; sq += v * v; }
    __shared__ float ls[256], lq[256];
    ls[tid] = s; lq[tid] = sq; __syncthreads();
    for (int off = 128; off > 0; off >>= 1) {
        if (tid < off) { ls[tid] += ls[tid + off]; lq[tid] += lq[tid + off]; }
        __syncthreads();
    }
    float inv = 1.0f / (ls[0] + EPS_SUM);
    if (tid == 0) kn[b] = sqrtf(lq[0]) * inv;   // ||key||/(sum+eps) == ||knorm||
    float* op = knorm + (size_t)b * DD;
    for (int i = tid; i < DD; i += 256) op[i] = kp[i] * inv;
}

// ---------------------------------------------------------------------------
// Kernel 1: dots[16,N] += knorm * mem^T  (f32 WMMA), mnsq[N] += ||row||^2.
// Double-buffered async global->LDS staging: stage t+1 copies overlap WMMA on
// stage t. grid = (N/16, KSPLIT), block = 256 (8 waves).
// ---------------------------------------------------------------------------
__global__ __launch_bounds__(256) void k1_dots(
        const float* __restrict__ mem, const float* __restrict__ knorm,
        float* __restrict__ dots, float* __restrict__ mnsq) {
    __shared__ float a_buf[2 * 16 * LDA];
    __shared__ float b_buf[2 * 16 * LDA];
    __shared__ float red[8 * 256];
    __shared__ float sqred[256];

    const int tid  = threadIdx.x;
    const int lane = tid & 31;
    const int wave = tid >> 5;
    const int hi   = lane >> 4;      // half-wave select
    const int lo   = lane & 15;
    const int n0   = blockIdx.x * 16;
    const int kbeg = blockIdx.y * KCHUNK;

    // staging mapping: 16 threads per tile row, float4 granularity (2 each)
    const int r = tid >> 4;          // tile row 0..15
    const int c = tid & 15;          // float4 lane within row
    const float* mrow = mem   + (size_t)(n0 + r) * DD;
    const float* arow = knorm + (size_t)r * DD;
    const uint32_t lb = lds_offset(b_buf) + (uint32_t)(r * LDA * 4);
    const uint32_t la = lds_offset(a_buf) + (uint32_t)(r * LDA * 4);
    const uint32_t bufstride = 16 * LDA * 4;

    // prologue: stage tile 0 into buffer 0
#pragma unroll
    for (int j = 0; j < 2; ++j) {
        int c4 = j * 16 + c;                       // coalesced 256B segments
        async_ld_b128(lb + c4 * 16, mrow + kbeg + c4 * 4);
        async_ld_b128(la + c4 * 16, arow + kbeg + c4 * 4);
    }
    wait_async0();
    __syncthreads();

    float sq = 0.f;
    v8f acc = {};
    for (int t = 0; t < NTILEK; ++t) {
        const int cur = t & 1;
        if (t + 1 < NTILEK) {                      // prefetch next stage
            const int knext = kbeg + (t + 1) * KB;
            const uint32_t boff = (cur ^ 1) * bufstride;
#pragma unroll
            for (int j = 0; j < 2; ++j) {
                int c4 = j * 16 + c;
                async_ld_b128(lb + boff + c4 * 16, mrow + knext + c4 * 4);
                async_ld_b128(la + boff + c4 * 16, arow + knext + c4 * 4);
            }
        }
        const float* at = a_buf + cur * (16 * LDA);
        const float* bt = b_buf + cur * (16 * LDA);
        const int kk0 = wave * (KB / 8);           // 16 K per wave per stage
#pragma unroll
        for (int kk = kk0; kk < kk0 + KB / 8; kk += 4) {
            v2f a, b;
            // A 16x4: lanes0-15 hold K={kk,kk+1}, lanes16-31 K={kk+2,kk+3}, M=lo
            a.x = at[lo * LDA + kk + 2 * hi];
            a.y = at[lo * LDA + kk + 2 * hi + 1];
            // B 4x16: VGPR0 rows K={kk+hi}, VGPR1 rows K={kk+2+hi}, Ncol=lo
            b.x = bt[lo * LDA + kk + hi];
            b.y = bt[lo * LDA + kk + 2 + hi];
            acc = wmma_f32(a, b, acc);
        }
        // memory-row sumsq from the staged B tile (conflict-free read-back)
#pragma unroll
        for (int j = 0; j < 2; ++j) {
            int c4 = j * 16 + c;
            float4 vb = *(const float4*)&bt[r * LDA + c4 * 4];
            sq += vb.x * vb.x + vb.y * vb.y + vb.z * vb.z + vb.w * vb.w;
        }
        wait_async0();                             // own prefetches landed
        __syncthreads();                           // everyone done reading cur
    }

    // cross-wave reduction of the 16x16 f32 tile
#pragma unroll
    for (int g = 0; g < 8; ++g) red[wave * 256 + lane * 8 + g] = acc[g];
    sqred[tid] = sq;
    __syncthreads();
    float tot = 0.f;
#pragma unroll
    for (int w2 = 0; w2 < 8; ++w2) tot += red[w2 * 256 + tid];
    // C/D layout: VGPR g, lane l -> M = g + (l>>4)*8, Ncol = l&15
    int g  = tid & 7;
    int l2 = tid >> 3;
    int m  = g + ((l2 >> 4) << 3);
    int n  = n0 + (l2 & 15);
    atomicAdd(&dots[m * NN + n], tot);
    if (tid < 16) {
        float ssum = 0.f;
        for (int i = 0; i < 16; ++i) ssum += sqred[tid * 16 + i];
        atomicAdd(&mnsq[n0 + tid], ssum);
    }
}

// ---------------------------------------------------------------------------
// Kernel 2: w[b,:] = softmax( tan( (dots/(kn*mn)) * pi/2 ) ) over N
// ---------------------------------------------------------------------------
__global__ __launch_bounds__(256) void k2_softmax(
        const float* __restrict__ dots, const float* __restrict__ kn,
        const float* __restrict__ mnsq, float* __restrict__ wout) {
    int b = blockIdx.x, tid = threadIdx.x;
    __shared__ float sh[256];
    float knb = fmaxf(kn[b], EPS_COS);
    float logit[8];
    float lmax = -3.4e38f;
#pragma unroll
    for (int i = 0; i < 8; ++i) {
        int n = tid + i * 256;
        float lg = -3.4e38f;
        if (n < NN) {
            float cosv = dots[b * NN + n] / (knb * fmaxf(sqrtf(mnsq[n]), EPS_COS));
            lg = tanf(cosv * PI_2F);
        }
        logit[i] = lg;
        lmax = fmaxf(lmax, lg);
    }
    sh[tid] = lmax; __syncthreads();
    for (int off = 128; off > 0; off >>= 1) {
        if (tid < off) sh[tid] = fmaxf(sh[tid], sh[tid + off]);
        __syncthreads();
    }
    float mx = sh[0]; __syncthreads();
    float lsum = 0.f;
#pragma unroll
    for (int i = 0; i < 8; ++i) {
        int n = tid + i * 256;
        if (n < NN) { float e = __expf(logit[i] - mx); wout[b * NN + n] = e; lsum += e; }
    }
    sh[tid] = lsum; __syncthreads();
    for (int off = 128; off > 0; off >>= 1) {
        if (tid < off) sh[tid] += sh[tid + off];
        __syncthreads();
    }
    float inv = 1.0f / sh[0];
#pragma unroll
    for (int i = 0; i < 8; ++i) {
        int n = tid + i * 256;
        if (n < NN) wout[b * NN + n] *= inv;
    }
}

// ---------------------------------------------------------------------------
// Kernel 3: out[16,D] = w[16,N] * mem[N,D]  (f32 WMMA, direct cacheline-exact
// global B loads; each half-wave consumes one 64B segment, adjacent waves the
// sibling half-line). grid = D/128, block = 256; wave owns 16 d-columns.
// ---------------------------------------------------------------------------
__global__ __launch_bounds__(256) void k3_readout(
        const float* __restrict__ mem, const float* __restrict__ wts,
        float* __restrict__ out) {
    const int tid  = threadIdx.x;
    const int lane = tid & 31;
    const int wave = tid >> 5;
    const int hi   = lane >> 4;
    const int lo   = lane & 15;
    const int dbase = blockIdx.x * 128 + wave * 16;

    v8f acc = {};
    const float* arow = wts + lo * NN;        // A row: M = b = lo
    for (int k0 = 0; k0 < NN; k0 += 4) {
        v2f a;
        a.x = arow[k0 + 2 * hi];
        a.y = arow[k0 + 2 * hi + 1];
        const float* mr0 = mem + (size_t)(k0 + hi) * DD + dbase;       // K = k0+hi
        const float* mr1 = mem + (size_t)(k0 + 2 + hi) * DD + dbase;   // K = k0+2+hi
        v2f b;
        b.x = mr0[lo];                        // cols dbase..dbase+15
        b.y = mr1[lo];
        acc = wmma_f32(a, b, acc);
    }
#pragma unroll
    for (int g = 0; g < 8; ++g) {
        int m = g + hi * 8;
        out[(size_t)m * DD + dbase + lo] = acc[g];
    }
}

// ---------------------------------------------------------------------------
// Host launch
// ---------------------------------------------------------------------------
extern "C" void kernel_launch(void* const* d_in, const int* in_sizes, int n_in,
                              void* d_out, int out_size, void* d_ws, size_t ws_size,
                              hipStream_t stream) {
    const float* key = (const float*)d_in[0];   // [16, 512, 7, 7]
    const float* mem = (const float*)d_in[1];   // [2000, 512, 7, 7]
    float* out = (float*)d_out;                 // [16, 512, 7, 7]

    float* ws    = (float*)d_ws;
    float* knorm = ws;                          // 16*25088
    float* dots  = knorm + (size_t)BB * DD;     // 16*2000
    float* wbuf  = dots + BB * NN;              // 16*2000
    float* kn    = wbuf + BB * NN;              // 16
    float* mnsq  = kn + BB;                     // 2000

    // zero the accumulation targets (dots, mnsq) + kn/wbuf region
    hipMemsetAsync(dots, 0, (size_t)(2 * BB * NN + BB + NN) * sizeof(float), stream);

    k0_normalize<<<BB, 256, 0, stream>>>(key, knorm, kn);
    k1_dots<<<dim3(NN / 16, KSPLIT), 256, 0, stream>>>(mem, knorm, dots, mnsq);
    k2_softmax<<<BB, 256, 0, stream>>>(dots, kn, mnsq, wbuf);
    k3_readout<<<DD / 128, 256, 0, stream>>>(mem, wbuf, out);
}